// NormAttention_38577396253110
// MI455X (gfx1250) — compile-verified
//
#include <hip/hip_runtime.h>

// ---------------------------------------------------------------------------
// NormAttention for MI455X (gfx1250): fp32 WMMA everywhere.
//   B=2, L=2048, E=512, H=8, hd=64
// GEMMs stage the shared X strip in LDS (320KB/WGP on CDNA5) so the hot loop
// is ds_load_b64 (A) + 3x global_load_b64 (B) + 3x v_wmma_f32_16x16x4_f32.
// ---------------------------------------------------------------------------

typedef float v2f __attribute__((ext_vector_type(2)));
typedef float v8f __attribute__((ext_vector_type(8)));

#define E_DIM   512
#define B_SZ    2
#define L_SEQ   2048
#define NHEAD   8
#define HD      64
#define NROWS   (L_SEQ * B_SZ)          // 4096 rows in (L,B,E) flat order
#define ROWSTR  (B_SZ * E_DIM)          // 1024: stride between consecutive l
#define XS_STR  516                     // padded LDS stride: 516%64=4 -> no bank conflicts

// ---- WMMA helpers ----------------------------------------------------------
// D = A(16x4 f32) x B(4x16 f32) + C(16x16 f32)
__device__ __forceinline__ v8f wmma4(v2f a, v2f b, v8f c) {
  return __builtin_amdgcn_wmma_f32_16x16x4_f32(
      /*neg_a=*/false, a, /*neg_b=*/false, b,
      /*c_mod=*/(short)0, c, /*reuse_a=*/false, /*reuse_b=*/false);
}

// A-fragment: A[m][k], m = lane&15, k = k0 + 2*(lane>>4) + {0,1}
__device__ __forceinline__ v2f load_a_frag(const float* src, int stride,
                                           int k0, int lane) {
  int m  = lane & 15;
  int kh = (lane >> 4) * 2;
  const float* p = src + m * stride + k0 + kh;
  v2f r; r.x = p[0]; r.y = p[1];
  return r;
}

// B-fragment where B[k][n] = S[k][n0+n] (S row-major, stride elements).
__device__ __forceinline__ v2f load_b_frag_n(const float* src, int stride,
                                             int k0, int n0, int lane) {
  int n  = lane & 15;
  int kh = (lane >> 4) * 2;
  const float* p = src + (k0 + kh) * stride + n0 + n;
  v2f r; r.x = p[0]; r.y = p[stride];
  return r;
}

// B-fragment where B[k][n] = T[n0+n][k] (i.e. B = T^T; T row-major).
__device__ __forceinline__ v2f load_b_frag_t(const float* src, int stride,
                                             int k0, int n0, int lane) {
  int n  = lane & 15;
  int kh = (lane >> 4) * 2;
  const float* p = src + (n0 + n) * stride + k0 + kh;
  v2f r; r.x = p[0]; r.y = p[1];
  return r;
}

// ---------------------------------------------------------------------------
// Kernel 1: fused QKV projection.
//   q = relu(X Wq^T + bq), k = relu(X Wk^T + bk), v = X Wv^T + bv
// One block per 16-row strip of X (256 blocks). The strip (16x512 fp32, 32KB)
// is staged in LDS once; 8 waves each sweep 4 N-tiles x {q,k,v}.
// ---------------------------------------------------------------------------
__global__ __launch_bounds__(256) void qkv_proj_kernel(
    const float* __restrict__ x,
    const float* __restrict__ Wq, const float* __restrict__ bq,
    const float* __restrict__ Wk, const float* __restrict__ bk,
    const float* __restrict__ Wv, const float* __restrict__ bv,
    float* __restrict__ q, float* __restrict__ k, float* __restrict__ v)
{
  __shared__ float Xs[16 * XS_STR];
  int t    = threadIdx.x;
  int lane = t & 31;
  int w    = t >> 5;                 // wave 0..7
  int m0   = blockIdx.x * 16;

  // stage X strip: 16 rows x 512 cols, 256 threads x 8 float4
  {
    int r = t >> 4;                  // 0..15
    int c = (t & 15) * 4;            // 0..60
#pragma unroll
    for (int p = 0; p < 8; ++p) {
      float4 f = *(const float4*)(x + (long)(m0 + r) * E_DIM + p * 64 + c);
      *(float4*)&Xs[r * XS_STR + p * 64 + c] = f;
    }
  }
  __syncthreads();

  for (int it = 0; it < 4; ++it) {
    int n0 = (w + 8 * it) * 16;      // waves cover all 32 N-tiles
    v8f aq = {}; v8f ak = {}; v8f av = {};
    for (int k0 = 0; k0 < E_DIM; k0 += 4) {
      v2f a   = load_a_frag(Xs, XS_STR, k0, lane);        // ds_load_b64
      v2f bqf = load_b_frag_t(Wq, E_DIM, k0, n0, lane);   // B = W^T
      v2f bkf = load_b_frag_t(Wk, E_DIM, k0, n0, lane);
      v2f bvf = load_b_frag_t(Wv, E_DIM, k0, n0, lane);
      aq = wmma4(a, bqf, aq);
      ak = wmma4(a, bkf, ak);
      av = wmma4(a, bvf, av);
    }
    int n   = n0 + (lane & 15);
    int mh  = 8 * (lane >> 4);
    float biq = bq[n], bik = bk[n], biv = bv[n];
#pragma unroll
    for (int i = 0; i < 8; ++i) {
      int m = m0 + mh + i;
      float fq = aq[i] + biq; fq = fq > 0.f ? fq : 0.f;
      float fk = ak[i] + bik; fk = fk > 0.f ? fk : 0.f;
      q[m * E_DIM + n] = fq;
      k[m * E_DIM + n] = fk;
      v[m * E_DIM + n] = av[i] + biv;
    }
  }
}

// ---------------------------------------------------------------------------
// Kernel 2: in-place L2 normalize each 512-long row of q and k.
// ---------------------------------------------------------------------------
__global__ __launch_bounds__(256) void l2norm_kernel(float* __restrict__ q,
                                                     float* __restrict__ k)
{
  __shared__ float red[256];
  int r = blockIdx.x;
  int t = threadIdx.x;
  float* bufs[2] = { q, k };
  for (int which = 0; which < 2; ++which) {
    float* p = bufs[which] + r * E_DIM;
    float a = p[t], b = p[t + 256];
    red[t] = a * a + b * b;
    __syncthreads();
    for (int s = 128; s > 0; s >>= 1) {
      if (t < s) red[t] += red[t + s];
      __syncthreads();
    }
    float scale = 1.0f / fmaxf(sqrtf(red[0]), 1e-12f);
    p[t]       = a * scale;
    p[t + 256] = b * scale;
    __syncthreads();
  }
}

// ---------------------------------------------------------------------------
// Kernel 3: chunked causal linear attention, one workgroup per head (16).
// LDS-resident KV state (64x64 f32). Per 16-row chunk:
//   P  = tril(Q K^T)            (wave 4)
//   O  = Q * KV_prev            (waves 0..3, one 16-col tile each)
//   O += P * V ; store          (waves 0..3)
//   KV += K^T * V               (all 8 waves, 2 tiles of 16x16 each)
// ---------------------------------------------------------------------------
__global__ __launch_bounds__(256) void linattn_kernel(
    const float* __restrict__ q, const float* __restrict__ k,
    const float* __restrict__ v, float* __restrict__ out)
{
  __shared__ float kvs[64 * 64];   // KV state, row = d (k-dim), col = m (v-dim)
  __shared__ float Qs[16 * 64];
  __shared__ float Ks[16 * 64];
  __shared__ float Vs[16 * 64];
  __shared__ float Ps[16 * 16];

  int head = blockIdx.x;                 // n = b*H + h
  int b    = head / NHEAD;
  int h    = head % NHEAD;
  int t    = threadIdx.x;
  int lane = t & 31;
  int w    = t >> 5;                     // wave id 0..7
  int colbase = b * E_DIM + h * HD;

  for (int i = t; i < 64 * 64; i += 256) kvs[i] = 0.f;
  __syncthreads();

  int ldr = t >> 4;                      // 0..15 row within chunk
  int ldc = (t & 15) * 4;                // 0..60 col (x4 floats)

  for (int c = 0; c < L_SEQ / 16; ++c) {
    int l0 = c * 16;

    // ---- cooperative load of Q/K/V chunk tiles (16x64 each) ----
    {
      long gidx = (long)(l0 + ldr) * ROWSTR + colbase + ldc;
      *(float4*)&Qs[ldr * 64 + ldc] = *(const float4*)(q + gidx);
      *(float4*)&Ks[ldr * 64 + ldc] = *(const float4*)(k + gidx);
      *(float4*)&Vs[ldr * 64 + ldc] = *(const float4*)(v + gidx);
    }
    __syncthreads();

    // ---- phase 2: O_partial = Q @ KV_prev (waves 0..3); P (wave 4) ----
    v8f acc = {};
    if (w < 4) {
      for (int k0 = 0; k0 < 64; k0 += 4) {
        v2f a  = load_a_frag(Qs, 64, k0, lane);
        v2f bf = load_b_frag_n(kvs, 64, k0, w * 16, lane);
        acc = wmma4(a, bf, acc);
      }
    } else if (w == 4) {
      v8f pr = {};
      for (int k0 = 0; k0 < 64; k0 += 4) {
        v2f a  = load_a_frag(Qs, 64, k0, lane);
        v2f bf = load_b_frag_t(Ks, 64, k0, 0, lane);   // B = K^T
        pr = wmma4(a, bf, pr);
      }
      int n  = lane & 15;
      int mh = 8 * (lane >> 4);
#pragma unroll
      for (int i = 0; i < 8; ++i) {
        int m = mh + i;
        Ps[m * 16 + n] = (n <= m) ? pr[i] : 0.f;       // causal mask, diag incl.
      }
    }
    __syncthreads();

    // ---- phase 3a: O += P @ V, store (waves 0..3) ----
    if (w < 4) {
      for (int k0 = 0; k0 < 16; k0 += 4) {
        v2f a  = load_a_frag(Ps, 16, k0, lane);
        v2f bf = load_b_frag_n(Vs, 64, k0, w * 16, lane);
        acc = wmma4(a, bf, acc);
      }
      int n  = lane & 15;
      int mh = 8 * (lane >> 4);
      float* o = out + colbase + w * 16 + n;
#pragma unroll
      for (int i = 0; i < 8; ++i)
        o[(long)(l0 + mh + i) * ROWSTR] = acc[i];
    }

    // ---- phase 3b: KV += K^T @ V ; wave w owns tiles 2w, 2w+1 ----
    for (int tt = 2 * w; tt < 2 * w + 2; ++tt) {
      int r  = tt >> 2;          // 16-row block of KV (k-dim)
      int cc = tt & 3;           // 16-col block of KV (v-dim)
      int n  = lane & 15;
      int mh = 8 * (lane >> 4);
      v8f ckv;
#pragma unroll
      for (int i = 0; i < 8; ++i)
        ckv[i] = kvs[(r * 16 + mh + i) * 64 + cc * 16 + n];
      for (int k0 = 0; k0 < 16; k0 += 4) {
        // A = (K chunk)^T tile: A[m][kk] = Ks[kk][r*16+m] -> same lane map as b_frag_n
        v2f a  = load_b_frag_n(Ks, 64, k0, r * 16, lane);
        v2f bf = load_b_frag_n(Vs, 64, k0, cc * 16, lane);
        ckv = wmma4(a, bf, ckv);
      }
#pragma unroll
      for (int i = 0; i < 8; ++i)
        kvs[(r * 16 + mh + i) * 64 + cc * 16 + n] = ckv[i];
    }
    __syncthreads();
  }
}

// ---------------------------------------------------------------------------
// Kernel 4: output projection  out = att @ Wo^T + bo (LDS-staged like kernel 1)
// ---------------------------------------------------------------------------
__global__ __launch_bounds__(256) void out_proj_kernel(
    const float* __restrict__ x, const float* __restrict__ Wo,
    const float* __restrict__ bo, float* __restrict__ out)
{
  __shared__ float Xs[16 * XS_STR];
  int t    = threadIdx.x;
  int lane = t & 31;
  int w    = t >> 5;
  int m0   = blockIdx.x * 16;

  {
    int r = t >> 4;
    int c = (t & 15) * 4;
#pragma unroll
    for (int p = 0; p < 8; ++p) {
      float4 f = *(const float4*)(x + (long)(m0 + r) * E_DIM + p * 64 + c);
      *(float4*)&Xs[r * XS_STR + p * 64 + c] = f;
    }
  }
  __syncthreads();

  for (int it = 0; it < 4; ++it) {
    int n0 = (w + 8 * it) * 16;
    v8f acc = {};
    for (int k0 = 0; k0 < E_DIM; k0 += 4) {
      v2f a  = load_a_frag(Xs, XS_STR, k0, lane);
      v2f bf = load_b_frag_t(Wo, E_DIM, k0, n0, lane);
      acc = wmma4(a, bf, acc);
    }
    int n  = n0 + (lane & 15);
    int mh = 8 * (lane >> 4);
    float bi = bo[n];
#pragma unroll
    for (int i = 0; i < 8; ++i)
      out[(m0 + mh + i) * E_DIM + n] = acc[i] + bi;
  }
}

// ---------------------------------------------------------------------------
extern "C" void kernel_launch(void* const* d_in, const int* in_sizes, int n_in,
                              void* d_out, int out_size, void* d_ws, size_t ws_size,
                              hipStream_t stream) {
  (void)in_sizes; (void)n_in; (void)out_size; (void)ws_size;
  const float* x  = (const float*)d_in[0];
  const float* Wq = (const float*)d_in[1];
  const float* bq = (const float*)d_in[2];
  const float* Wk = (const float*)d_in[3];
  const float* bk = (const float*)d_in[4];
  const float* Wv = (const float*)d_in[5];
  const float* bv = (const float*)d_in[6];
  const float* Wo = (const float*)d_in[7];
  const float* bo = (const float*)d_in[8];
  float* out = (float*)d_out;

  float* qw = (float*)d_ws;                 // 4096x512 each
  float* kw = qw + (size_t)NROWS * E_DIM;
  float* vw = kw + (size_t)NROWS * E_DIM;
  float* aw = vw + (size_t)NROWS * E_DIM;

  const int strips = NROWS / 16;            // 256 blocks, one 16-row strip each

  qkv_proj_kernel<<<strips, 256, 0, stream>>>(x, Wq, bq, Wk, bk, Wv, bv, qw, kw, vw);
  l2norm_kernel  <<<NROWS, 256, 0, stream>>>(qw, kw);
  linattn_kernel <<<B_SZ * NHEAD, 256, 0, stream>>>(qw, kw, vw, aw);
  out_proj_kernel<<<strips, 256, 0, stream>>>(aw, Wo, bo, out);
}